// _Wrapper_66254165508467
// MI455X (gfx1250) — compile-verified
//
#include <hip/hip_runtime.h>
#include <stdint.h>

#define NUM_CLASSES 90
#define NUM_ANCHORS 9
#define K_TOP 5000
#define BATCH 16
#define NBINS 2048           // top 11 bits of orderable key
#define TILE 4096            // floats per TDM tile (16 KB)
#define TILES_PER_WG 16      // 64 K elements per workgroup chunk
#define SORT_N 8192          // next pow2 >= K_TOP

typedef unsigned int u32x4 __attribute__((ext_vector_type(4)));
typedef int i32x4 __attribute__((ext_vector_type(4)));
typedef int i32x8 __attribute__((ext_vector_type(8)));

// ---------- helpers ----------

// LDS byte offset of a __shared__ object (generic -> AS3 -> int)
__device__ __forceinline__ unsigned lds_byte_offset(const void* p) {
  return (unsigned)(unsigned long long)(__attribute__((address_space(3))) const void*)p;
}

// monotonic float -> uint transform (larger float => larger key)
__device__ __forceinline__ unsigned order_key(float f) {
  unsigned bits = __float_as_uint(f);
  return (bits & 0x80000000u) ? ~bits : (bits | 0x80000000u);
}
__device__ __forceinline__ float key_to_float(unsigned key) {
  unsigned bits = (key & 0x80000000u) ? (key ^ 0x80000000u) : ~key;
  return __uint_as_float(bits);
}

// Issue one TDM DMA: `count` floats from gsrc to the LDS byte offset lds_byte, tile_dim0=TILE
// (tail elements beyond tensor_dim0=count zero-fill via TDM OOB semantics).
// NOTE: intentionally NOT guarded by __has_builtin -- a missing/mis-aritied
// builtin must be a hard compile error so the feedback loop can see it.
__device__ __forceinline__ void tdm_load_tile(const float* gsrc, unsigned lds_byte,
                                              unsigned count) {
  unsigned long long ga = (unsigned long long)(uintptr_t)gsrc;
  // D# group0: count=1(valid) | lds_addr | global_addr[56:0] | type=2
  u32x4 g0 = { 1u,
               lds_byte,
               (unsigned)ga,
               (unsigned)((ga >> 32) & 0x1FFFFFFu) | (2u << 30) };
  // D# group1: data_size=4B; tensor_dim0=count; tensor_dim1=1;
  //            tile_dim0=TILE; tile_dim1=1; tensor_dim0_stride=count
  i32x8 g1 = { (int)0x20000u,
               (int)((count & 0xFFFFu) << 16),
               (int)(((count >> 16) & 0xFFFFu) | (1u << 16)),
               (int)((unsigned)TILE << 16),
               1,
               (int)count,
               0, 0 };
  i32x4 z4 = {0, 0, 0, 0};
#if __clang_major__ >= 23
  i32x8 z8 = {0, 0, 0, 0, 0, 0, 0, 0};
  __builtin_amdgcn_tensor_load_to_lds(g0, g1, z4, z4, z8, 0);
#else
  __builtin_amdgcn_tensor_load_to_lds(g0, g1, z4, z4, 0);
#endif
}

// ---------- kernel 0: zero hist/thresh/counters ----------
__global__ void init_kernel(unsigned* __restrict__ ws, int n) {
  int i = blockIdx.x * blockDim.x + threadIdx.x;
  if (i < n) ws[i] = 0u;
}

// ---------- kernel 1: per-batch 2048-bin histogram, TDM double-buffered ----------
__global__ void __launch_bounds__(256)
hist_kernel(const float* __restrict__ src, unsigned* __restrict__ hist,
            int perBatch) {
  __shared__ float buf[2][TILE];      // 32 KB double buffer
  __shared__ unsigned lhist[NBINS];   // 8 KB private histogram
  const int b = blockIdx.y;
  const int chunkBase = blockIdx.x * (TILE * TILES_PER_WG);
  if (chunkBase >= perBatch) return;
  const float* sb = src + (size_t)b * perBatch;

  int remain = perBatch - chunkBase;
  int nTiles = (remain + TILE - 1) / TILE;
  if (nTiles > TILES_PER_WG) nTiles = TILES_PER_WG;

  for (int i = threadIdx.x; i < NBINS; i += blockDim.x) lhist[i] = 0u;

  const bool issuer = (threadIdx.x < 32);  // wave 0 only: TDM ignores EXEC
  if (issuer) {
    unsigned c0 = (unsigned)min(TILE, perBatch - chunkBase);
    tdm_load_tile(sb + chunkBase, lds_byte_offset(&buf[0][0]), c0);
  }
  for (int t = 0; t < nTiles; ++t) {
    if (issuer) {
      if (t + 1 < nTiles) {
        int ns = chunkBase + (t + 1) * TILE;
        unsigned c = (unsigned)min(TILE, perBatch - ns);
        tdm_load_tile(sb + ns, lds_byte_offset(&buf[(t + 1) & 1][0]), c);
        __builtin_amdgcn_s_wait_tensorcnt(1);  // oldest DMA (tile t) done
      } else {
        __builtin_amdgcn_s_wait_tensorcnt(0);
      }
    }
    __syncthreads();                         // tile t visible to all waves
    int start = chunkBase + t * TILE;
    int cnt = min(TILE, perBatch - start);   // always a multiple of 4
    const float4* lb4 = (const float4*)buf[t & 1];
    int cnt4 = cnt >> 2;
    for (int i = threadIdx.x; i < cnt4; i += blockDim.x) {
      float4 f4 = lb4[i];                    // ds_load_b128
      atomicAdd(&lhist[order_key(f4.x) >> 21], 1u);
      atomicAdd(&lhist[order_key(f4.y) >> 21], 1u);
      atomicAdd(&lhist[order_key(f4.z) >> 21], 1u);
      atomicAdd(&lhist[order_key(f4.w) >> 21], 1u);
    }
    __syncthreads();                         // done with buf[t&1] before reuse
  }
  unsigned* hb = hist + (size_t)b * NBINS;
  for (int i = threadIdx.x; i < NBINS; i += blockDim.x) {
    unsigned v = lhist[i];
    if (v) atomicAdd(&hb[i], v);
  }
}

// ---------- kernel 2: suffix scan -> threshold bin + residual need ----------
__global__ void scan_kernel(const unsigned* __restrict__ hist,
                            unsigned* __restrict__ thresh) {
  int b = blockIdx.x;
  if (threadIdx.x == 0) {
    unsigned cum = 0, bin = 0, need = K_TOP;
    for (int i = NBINS - 1; i >= 0; --i) {
      unsigned c = hist[(size_t)b * NBINS + i];
      if (cum + c >= (unsigned)K_TOP) { bin = (unsigned)i; need = K_TOP - cum; break; }
      cum += c;
    }
    thresh[b * 4 + 0] = bin;
    thresh[b * 4 + 1] = need;
  }
}

// ---------- kernel 3: compaction (2nd stream, b128 loads + prefetch) ----------
__global__ void __launch_bounds__(256)
compact_kernel(const float4* __restrict__ src, const unsigned* __restrict__ thresh,
               unsigned* __restrict__ cntMain, unsigned* __restrict__ cntBin,
               unsigned* __restrict__ keys, unsigned* __restrict__ idxs,
               int vecPerBatch, int logHW, int anchorOff) {
  const int b = blockIdx.y;
  const unsigned tbin = thresh[b * 4 + 0];
  const unsigned need = thresh[b * 4 + 1];
  const float4* sb = src + (size_t)b * vecPerBatch;
  const int stride = gridDim.x * blockDim.x;
  const int hwMask = (1 << logHW) - 1;
  for (int v = blockIdx.x * blockDim.x + threadIdx.x; v < vecPerBatch; v += stride) {
    if (v + stride < vecPerBatch) __builtin_prefetch(sb + v + stride, 0, 1);
    float4 f4 = sb[v];
    float fs[4] = {f4.x, f4.y, f4.z, f4.w};
#pragma unroll
    for (int j = 0; j < 4; ++j) {
      unsigned key = order_key(fs[j]);
      unsigned bin = key >> 21;
      if (bin < tbin) continue;
      bool take = (bin > tbin);
      if (!take) take = (atomicAdd(&cntBin[b], 1u) < need);  // ties: admit `need`
      if (take) {
        unsigned pos = atomicAdd(&cntMain[b], 1u);
        if (pos < K_TOP) {
          int e = v * 4 + j;                 // NCHW element index within batch
          int ch = e >> logHW;               // channel = a*90 + c
          int pix = e & hwMask;              // y*W + x
          int a = ch / NUM_CLASSES;
          int c = ch - a * NUM_CLASSES;
          unsigned n = (unsigned)(anchorOff + pix * NUM_ANCHORS + a);
          keys[(size_t)b * K_TOP + pos] = key;
          idxs[(size_t)b * K_TOP + pos] = n * (unsigned)NUM_CLASSES + (unsigned)c;
        }
      }
    }
  }
}

// ---------- kernel 4: per-batch bitonic sort (descending) in LDS ----------
__global__ void __launch_bounds__(1024)
sort_kernel(const unsigned* __restrict__ keys, unsigned* __restrict__ skeys,
            unsigned* __restrict__ order) {
  __shared__ unsigned sk[SORT_N];        // 32 KB keys
  __shared__ unsigned short si[SORT_N];  // 16 KB payload (candidate position)
  int b = blockIdx.x;
  for (int i = threadIdx.x; i < SORT_N; i += blockDim.x) {
    sk[i] = (i < K_TOP) ? keys[(size_t)b * K_TOP + i] : 0u;  // pad = minimum key
    si[i] = (unsigned short)i;
  }
  __syncthreads();
  for (int k = 2; k <= SORT_N; k <<= 1) {
    for (int j = k >> 1; j > 0; j >>= 1) {
      for (int i = threadIdx.x; i < SORT_N; i += blockDim.x) {
        int ixj = i ^ j;
        if (ixj > i) {
          unsigned a = sk[i], c = sk[ixj];
          bool descSeg = ((i & k) == 0);
          if (descSeg ? (a < c) : (a > c)) {
            sk[i] = c; sk[ixj] = a;
            unsigned short t = si[i]; si[i] = si[ixj]; si[ixj] = t;
          }
        }
      }
      __syncthreads();
    }
  }
  for (int i = threadIdx.x; i < K_TOP; i += blockDim.x) {
    skeys[(size_t)b * K_TOP + i] = sk[i];
    order[(size_t)b * K_TOP + i] = (unsigned)si[i];
  }
}

// ---------- kernel 5: decode + box gather + write outputs ----------
__global__ void __launch_bounds__(256)
output_kernel(const unsigned* __restrict__ skeys, const unsigned* __restrict__ order,
              const unsigned* __restrict__ idxs,
              const float* __restrict__ b0, const float* __restrict__ b1,
              const float* __restrict__ b2, const float* __restrict__ b3,
              const float* __restrict__ b4, float* __restrict__ out) {
  int b = blockIdx.y;
  int r = blockIdx.x * blockDim.x + threadIdx.x;
  if (r >= K_TOP) return;
  size_t p = (size_t)b * K_TOP + r;
  unsigned key = skeys[p];
  unsigned flat = idxs[(size_t)b * K_TOP + order[p]];
  float score = key_to_float(key);
  unsigned n = flat / NUM_CLASSES;
  unsigned cls = flat - n * NUM_CLASSES;
  const float* bp; unsigned off, hw;
  if (n < 36864u)      { bp = b0; off = 0u;     hw = 4096u; }
  else if (n < 46080u) { bp = b1; off = 36864u; hw = 1024u; }
  else if (n < 48384u) { bp = b2; off = 46080u; hw = 256u;  }
  else if (n < 48960u) { bp = b3; off = 48384u; hw = 64u;   }
  else                 { bp = b4; off = 48960u; hw = 16u;   }
  unsigned r2 = n - off;
  unsigned pix = r2 / NUM_ANCHORS;
  unsigned a = r2 - pix * NUM_ANCHORS;
  const float* bb = bp + (size_t)b * (NUM_ANCHORS * 4) * hw;

  out[p] = score;                                        // cls_topk [B,K,1]
  size_t boxBase = (size_t)BATCH * K_TOP + p * 4;        // box_topk [B,K,4]
#pragma unroll
  for (int j = 0; j < 4; ++j)
    out[boxBase + j] = bb[(size_t)(a * 4 + j) * hw + pix];
  out[(size_t)BATCH * K_TOP * 5 + p] = (float)n;         // indices [B,K]
  out[(size_t)BATCH * K_TOP * 6 + p] = (float)cls;       // classes [B,K]
}

// ---------- host launch ----------
extern "C" void kernel_launch(void* const* d_in, const int* in_sizes, int n_in,
                              void* d_out, int out_size, void* d_ws, size_t ws_size,
                              hipStream_t stream) {
  (void)in_sizes; (void)n_in; (void)out_size; (void)ws_size;
  const float* cls[5];
  const float* box[5];
  for (int i = 0; i < 5; ++i) {
    cls[i] = (const float*)d_in[i];
    box[i] = (const float*)d_in[5 + i];
  }
  unsigned* ws      = (unsigned*)d_ws;
  unsigned* hist    = ws;                          // 16*2048
  unsigned* thresh  = hist + BATCH * NBINS;        // 16*4
  unsigned* cntMain = thresh + BATCH * 4;          // 16
  unsigned* cntBin  = cntMain + BATCH;             // 16
  unsigned* keys    = cntBin + BATCH;              // 16*5000
  unsigned* idxs    = keys + BATCH * K_TOP;        // 16*5000
  unsigned* skeys   = idxs + BATCH * K_TOP;        // 16*5000
  unsigned* order   = skeys + BATCH * K_TOP;       // 16*5000

  static const int HWs[5]  = {4096, 1024, 256, 64, 16};
  static const int LOG[5]  = {12, 10, 8, 6, 4};
  static const int AOFF[5] = {0, 36864, 46080, 48384, 48960};

  int zeroN = BATCH * NBINS + BATCH * 4 + BATCH * 2;
  init_kernel<<<(zeroN + 255) / 256, 256, 0, stream>>>(ws, zeroN);

  for (int l = 0; l < 5; ++l) {
    int perBatch = NUM_ANCHORS * NUM_CLASSES * HWs[l];
    int chunks = (perBatch + TILE * TILES_PER_WG - 1) / (TILE * TILES_PER_WG);
    hist_kernel<<<dim3(chunks, BATCH), 256, 0, stream>>>(cls[l], hist, perBatch);
  }
  scan_kernel<<<BATCH, 32, 0, stream>>>(hist, thresh);
  for (int l = 0; l < 5; ++l) {
    int vec = (NUM_ANCHORS * NUM_CLASSES * HWs[l]) / 4;
    int blocks = (vec + 255) / 256;
    if (blocks > 256) blocks = 256;
    compact_kernel<<<dim3(blocks, BATCH), 256, 0, stream>>>(
        (const float4*)cls[l], thresh, cntMain, cntBin, keys, idxs,
        vec, LOG[l], AOFF[l]);
  }
  sort_kernel<<<BATCH, 1024, 0, stream>>>(keys, skeys, order);
  output_kernel<<<dim3((K_TOP + 255) / 256, BATCH), 256, 0, stream>>>(
      skeys, order, idxs, box[0], box[1], box[2], box[3], box[4], (float*)d_out);
}